// DotAttention_52501680226809
// MI455X (gfx1250) — compile-verified
//
#include <hip/hip_runtime.h>

typedef float v2f __attribute__((ext_vector_type(2)));
typedef float v8f __attribute__((ext_vector_type(8)));
typedef int   v4i __attribute__((ext_vector_type(4)));

// Probe for gfx1250 async global->LDS path; fall back to sync staging if the
// toolchain doesn't declare the builtins (compile must stay green).
#if defined(__has_builtin)
#if __has_builtin(__builtin_amdgcn_global_load_async_to_lds_b128) && \
    __has_builtin(__builtin_amdgcn_s_wait_asynccnt)
#define USE_ASYNC_LDS 1
#endif
#endif
#ifndef USE_ASYNC_LDS
#define USE_ASYNC_LDS 0
#endif

#define B_DIM 64
#define S_DIM 2048
#define H_DIM 1024
#define NCHUNK 8
#define ROWS_PER_CHUNK (S_DIM / NCHUNK)        // 256
#define GROUPS_PER_CHUNK (ROWS_PER_CHUNK / 16) // 16
#define TPAD 1028                              // 1024 + 4 pad -> conflict-free LDS

#if USE_ASYNC_LDS
// Builtin signature (from hipcc diagnostic): param0 is v4i in the global
// (__device__ / addrspace(1)) AS, non-const; param1 is the LDS-side pointer.
typedef __attribute__((address_space(1))) v4i* gvec_t;
typedef __attribute__((address_space(3))) v4i* lvec_t;
#endif

// Issue the fill of one 16x1024 f32 tile (64 KB). Async path: 16 per-thread
// GLOBAL_LOAD_ASYNC_TO_LDS_B128 ops (ASYNCcnt). Sync path: b128 load + ds_store.
__device__ __forceinline__ void issue_tile_fill(float (*dst)[TPAD],
                                                const float* __restrict__ src,
                                                int t) {
#if USE_ASYNC_LDS
#pragma unroll
    for (int j = 0; j < 16; ++j) {
        gvec_t g = (gvec_t)(src + (size_t)j * H_DIM + t * 4);
        lvec_t l = (lvec_t)&dst[j][t * 4];
        __builtin_amdgcn_global_load_async_to_lds_b128(g, l, 0, 0);
    }
#else
#pragma unroll
    for (int j = 0; j < 16; ++j) {
        const float4* s = (const float4*)(src + (size_t)j * H_DIM);
        *(float4*)&dst[j][t * 4] = s[t];
    }
#endif
}

// ---------------------------------------------------------------------------
// Kernel 1: flash-style partial attention over one (batch, s-chunk).
// Double-buffered LDS tiles: prefetch group g+1 (async) while computing g.
//   - scores via V_WMMA_F32_16X16X4_F32 accumulation over H
//   - online softmax (running m, l) + context accumulated from the LDS tile
//   - raw scores stashed into the attn region of d_out (normalized later)
// ---------------------------------------------------------------------------
__global__ __launch_bounds__(256)
void attn_flash_partial(const float* __restrict__ dec_g,
                        const float* __restrict__ enc_g,
                        float* __restrict__ attn_raw,   // d_out + B*H : raw scores
                        float* __restrict__ pm,
                        float* __restrict__ pl,
                        float* __restrict__ pacc) {
    __shared__ float dec[H_DIM];
    __shared__ float tile[2][16][TPAD];
    __shared__ float red[16];

    const int b     = blockIdx.x;
    const int chunk = blockIdx.y;
    const int t     = threadIdx.x;
    const int wave  = t >> 5;
    const int lane  = t & 31;
    const int lm    = lane & 15;   // row (A) / column (D) selector
    const int hi    = lane >> 4;   // K-half selector per ISA A layout

    // Load decoder row (4 KB) cooperatively, float4-coalesced.
    {
        const float4* src = (const float4*)(dec_g + (size_t)b * H_DIM);
        ((float4*)dec)[t] = src[t];
    }

    const float* ebase  = enc_g + (size_t)b * S_DIM * H_DIM;
    const int    s_base = chunk * ROWS_PER_CHUNK;

    // Prefetch group 0 into buffer 0.
    issue_tile_fill(tile[0], ebase + (size_t)s_base * H_DIM, t);

    float m = -3.4e38f;
    float l = 0.f;
    float acc0 = 0.f, acc1 = 0.f, acc2 = 0.f, acc3 = 0.f;

    for (int g = 0; g < GROUPS_PER_CHUNK; ++g) {
        const int cur = g & 1;
        const int nxt = cur ^ 1;
        const int s0  = s_base + g * 16;

        if (t < 16) red[t] = 0.f;

        if (g + 1 < GROUPS_PER_CHUNK) {
            // Prefetch next tile into the other buffer (safe: everyone passed
            // the end-of-previous-iteration barrier), then wait for the
            // *current* tile only: async loads complete in order, so <=16
            // outstanding means group g's 16 loads have landed.
            issue_tile_fill(tile[nxt], ebase + (size_t)(s0 + 16) * H_DIM, t);
#if USE_ASYNC_LDS
            __builtin_amdgcn_s_wait_asynccnt(16);
#endif
        } else {
#if USE_ASYNC_LDS
            __builtin_amdgcn_s_wait_asynccnt(0);
#endif
        }
        __syncthreads();   // tile[cur] visible to all waves; red zeroed

        // WMMA partial scores. Wave w covers h in [w*128, w*128+128).
        // A (16x4 f32): lane lm holds row lm, K = {0,1} (hi=0) or {2,3} (hi=1).
        // B (4x16 f32): decoder chunk replicated across all 16 columns, so
        // every column of D holds the same partial dot product.
        v8f c = {0.f, 0.f, 0.f, 0.f, 0.f, 0.f, 0.f, 0.f};
        const int colsel = 2 * hi;
#pragma unroll
        for (int kk = 0; kk < 32; ++kk) {
            const int k = wave * 128 + kk * 4 + colsel;
            v2f a  = *(const v2f*)&tile[cur][lm][k]; // banks (4*lm+k)%64: clean
            v2f bb = *(const v2f*)&dec[k];           // broadcast
            c = __builtin_amdgcn_wmma_f32_16x16x4_f32(false, a, false, bb,
                                                      (short)0, c, false, false);
        }
        // D layout: lane (N=lm), VGPR v -> M = v + 8*hi. Column-0 lanes carry
        // all 16 row results; reduce the 8 per-wave partials via ds_add_f32.
        if (lm == 0) {
#pragma unroll
            for (int v = 0; v < 8; ++v)
                atomicAdd(&red[hi * 8 + v], c[v]);
        }
        __syncthreads();

        // Online softmax update (identical in every thread).
        float newm = m;
#pragma unroll
        for (int i = 0; i < 16; ++i) newm = fmaxf(newm, red[i]);
        const float scale = __expf(m - newm);
        float p[16];
        float psum = 0.f;
#pragma unroll
        for (int i = 0; i < 16; ++i) { p[i] = __expf(red[i] - newm); psum += p[i]; }
        l = l * scale + psum;
        m = newm;

        // Context accumulation straight from the LDS tile (no second HBM pass).
        // Thread t owns columns t, t+256, t+512, t+768.
        acc0 *= scale; acc1 *= scale; acc2 *= scale; acc3 *= scale;
#pragma unroll
        for (int i = 0; i < 16; ++i) {
            acc0 = fmaf(p[i], tile[cur][i][t],       acc0);
            acc1 = fmaf(p[i], tile[cur][i][t + 256], acc1);
            acc2 = fmaf(p[i], tile[cur][i][t + 512], acc2);
            acc3 = fmaf(p[i], tile[cur][i][t + 768], acc3);
        }

        if (t < 16) attn_raw[(size_t)b * S_DIM + s0 + t] = red[t];
        __syncthreads();   // tile[cur]/red free for reuse next iteration
    }

    // Emit per-chunk partials.
    const int pidx = b * NCHUNK + chunk;
    if (t == 0) { pm[pidx] = m; pl[pidx] = l; }
    float* pa = pacc + (size_t)pidx * H_DIM;
    pa[t]       = acc0;
    pa[t + 256] = acc1;
    pa[t + 512] = acc2;
    pa[t + 768] = acc3;
}

// ---------------------------------------------------------------------------
// Kernel 2: combine the 8 per-chunk partials of each batch, write final
// context, and normalize the stashed raw scores into attention weights.
// ---------------------------------------------------------------------------
__global__ __launch_bounds__(256)
void attn_combine(const float* __restrict__ pm,
                  const float* __restrict__ pl,
                  const float* __restrict__ pacc,
                  float* __restrict__ ctx_out,
                  float* __restrict__ attn) {
    const int b = blockIdx.x;
    const int t = threadIdx.x;

    float mc[NCHUNK], lc[NCHUNK];
    float M = -3.4e38f;
#pragma unroll
    for (int c = 0; c < NCHUNK; ++c) {
        mc[c] = pm[b * NCHUNK + c];
        lc[c] = pl[b * NCHUNK + c];
        M = fmaxf(M, mc[c]);
    }
    float w[NCHUNK];
    float L = 0.f;
#pragma unroll
    for (int c = 0; c < NCHUNK; ++c) {
        w[c] = __expf(mc[c] - M);
        L = fmaf(lc[c], w[c], L);
    }
    const float invL = 1.f / L;

#pragma unroll
    for (int j = 0; j < 4; ++j) {
        const int col = t + j * 256;
        float s = 0.f;
#pragma unroll
        for (int c = 0; c < NCHUNK; ++c)
            s = fmaf(w[c], pacc[(size_t)(b * NCHUNK + c) * H_DIM + col], s);
        ctx_out[(size_t)b * H_DIM + col] = s * invL;
    }

    for (int s = t; s < S_DIM; s += 256) {
        const float raw = attn[(size_t)b * S_DIM + s];
        attn[(size_t)b * S_DIM + s] = __expf(raw - M) * invL;
    }
}

// ---------------------------------------------------------------------------
extern "C" void kernel_launch(void* const* d_in, const int* in_sizes, int n_in,
                              void* d_out, int out_size, void* d_ws, size_t ws_size,
                              hipStream_t stream) {
    (void)in_sizes; (void)n_in; (void)out_size; (void)ws_size;

    const float* dec = (const float*)d_in[0];   // [64, 1024] f32
    const float* enc = (const float*)d_in[1];   // [64, 2048, 1024] f32

    float* ctx  = (float*)d_out;                         // [64, 1024]
    float* attn = (float*)d_out + (size_t)B_DIM * H_DIM; // [64, 2048]

    float* pm   = (float*)d_ws;                 // [B*NCHUNK]
    float* pl   = pm + B_DIM * NCHUNK;          // [B*NCHUNK]
    float* pacc = pl + B_DIM * NCHUNK;          // [B*NCHUNK, H]  (~2 MB total)

    attn_flash_partial<<<dim3(B_DIM, NCHUNK), 256, 0, stream>>>(
        dec, enc, attn, pm, pl, pacc);
    attn_combine<<<B_DIM, 256, 0, stream>>>(pm, pl, pacc, ctx, attn);
}